// SamplingAttention_RA_53850299957573
// MI455X (gfx1250) — compile-verified
//
#include <hip/hip_runtime.h>
#include <hip/hip_bf16.h>
#include <math.h>

typedef __attribute__((ext_vector_type(2))) float v2f;
typedef __attribute__((ext_vector_type(8))) float v8f;

#define D_MODEL  256
#define N_BATCH  8
#define N_QUERY  300
#define N_HEADS  8
#define N_LEVELS 4
#define N_POINTS 4
#define HEAD_DIM 32
#define S_TOTAL  21760   // 128*128 + 64*64 + 32*32 + 16*16

#define LDS_STRIDE 260   // 256 + 4 pad: bank = (4*r + k) % 64 -> conflict-free frags

// ---------------------------------------------------------------------------
// GEMM-NT via V_WMMA_F32_16X16X4_F32, K = Ncols = 256 fixed:
//   C[m,n] = sum_k A[m,k] * B[n,k] + bias[n]   (optionally zeroed by mask[m])
// One block (8 waves) computes a full 16x256 output stripe:
//   - A 16x256 tile staged once in LDS (shared by all 8 waves)
//   - each wave owns two adjacent 16x16 column tiles, reusing its A fragment
// Per K-step per wave: 1 ds_load_b64 (A) + 2 global_load_b64 (B) + 2 WMMAs.
// ---------------------------------------------------------------------------
__global__ void __launch_bounds__(256) wmma_gemm_nt_kernel(
    const float* __restrict__ A, const float* __restrict__ Bw,
    const float* __restrict__ bias, float* __restrict__ C,
    int Mrows, const unsigned char* __restrict__ mask)
{
    const int K = D_MODEL, Ncols = D_MODEL;
    __shared__ float lds_a[16 * LDS_STRIDE];

    const int tileRow = blockIdx.x << 4;

    // cooperative A-tile stage: 256 threads x 16 floats (4x float4 each)
    {
        const int t    = threadIdx.x;
        const int row  = t >> 4;
        const int col0 = (t & 15) << 4;
        const float* src = A + (size_t)(tileRow + row) * K + col0;
        float*       dst = lds_a + row * LDS_STRIDE + col0;
#pragma unroll
        for (int i = 0; i < 4; ++i) {
            *(float4*)(dst + i * 4) = *(const float4*)(src + i * 4);
        }
    }
    __syncthreads();

    const int lane = threadIdx.x & 31;
    const int wave = threadIdx.x >> 5;
    const int r    = lane & 15;
    const int kg   = (lane >> 4) << 1;          // 0 or 2
    const int tileCol0 = wave << 5;             // two 16-col tiles per wave

    const float* aLds = lds_a + r * LDS_STRIDE + kg;
    const float* b0   = Bw + (size_t)(tileCol0 + r)      * K + kg;
    const float* b1   = Bw + (size_t)(tileCol0 + 16 + r) * K + kg;

    v8f acc0 = {}, acc1 = {};
#pragma unroll 8
    for (int k0 = 0; k0 < K; k0 += 4) {
        v2f a   = *(const v2f*)(aLds + k0);
        v2f bb0 = *(const v2f*)(b0 + k0);
        v2f bb1 = *(const v2f*)(b1 + k0);
        // (neg_a, A, neg_b, B, c_mod, C, reuse_a, reuse_b)
        acc0 = __builtin_amdgcn_wmma_f32_16x16x4_f32(
            false, a, false, bb0, (short)0, acc0, false, false);
        acc1 = __builtin_amdgcn_wmma_f32_16x16x4_f32(
            false, a, false, bb1, (short)0, acc1, false, false);
    }

    const float bc0   = bias[tileCol0 + r];
    const float bc1   = bias[tileCol0 + 16 + r];
    const int   rowHi = (lane >> 4) << 3;       // 0 or 8
#pragma unroll
    for (int i = 0; i < 8; ++i) {
        const int  row = tileRow + rowHi + i;
        const bool mz  = (mask != nullptr) && mask[row];
        float v0 = mz ? 0.0f : acc0[i] + bc0;
        float v1 = mz ? 0.0f : acc1[i] + bc1;
        float* crow = C + (size_t)row * Ncols;
        crow[tileCol0 + r]      = v0;
        crow[tileCol0 + 16 + r] = v1;
    }
}

// ---------------------------------------------------------------------------
// Query projections: wgt logits (NQ x 128) and sampling offsets (NQ x 256).
// One thread per output element; K=256 dot product with float4 loads.
// ---------------------------------------------------------------------------
__global__ void __launch_bounds__(256) qproj_kernel(
    const float* __restrict__ query,                      // (NQ, 256)
    const float* __restrict__ wgt_w, const float* __restrict__ wgt_b,
    const float* __restrict__ loc_w, const float* __restrict__ loc_b,
    float* __restrict__ wgt_raw,                          // (NQ, 128)
    float* __restrict__ locs)                             // (NQ, 256)
{
    const int NQ = N_BATCH * N_QUERY;
    int t = blockIdx.x * 256 + threadIdx.x;
    if (t >= NQ * 384) return;
    int row = t / 384;
    int j   = t - row * 384;

    const float4* q4 = reinterpret_cast<const float4*>(query + (size_t)row * D_MODEL);
    const float4* w4;
    float  acc;
    float* dst;
    if (j < 128) {
        w4  = reinterpret_cast<const float4*>(wgt_w + (size_t)j * D_MODEL);
        acc = wgt_b[j];
        dst = wgt_raw + (size_t)row * 128 + j;
    } else {
        const int j2 = j - 128;
        w4  = reinterpret_cast<const float4*>(loc_w + (size_t)j2 * D_MODEL);
        acc = loc_b[j2];
        dst = locs + (size_t)row * 256 + j2;
    }
#pragma unroll 8
    for (int k = 0; k < D_MODEL / 4; ++k) {
        const float4 q = q4[k], w = w4[k];
        acc += q.x * w.x + q.y * w.y + q.z * w.z + q.w * w.w;
    }
    *dst = acc;
}

// ---------------------------------------------------------------------------
// Sampling + softmax + weighted accumulation.
// One wave per (n, l, m); lane = channel within head (HEAD_DIM == 32 == wave32).
// All grid math is wave-uniform; corner gathers are coalesced 128B bursts.
// ---------------------------------------------------------------------------
__global__ void __launch_bounds__(256) msda_sample_kernel(
    const float* __restrict__ value,      // (N, S, C) in ws
    const float* __restrict__ wgt_raw,    // (N*L*M, 16) logits
    const float* __restrict__ locs,       // (N*L, 256) = (m,p,f,xy)
    const float* __restrict__ refp,       // (N, L, Fl, 4)
    float* __restrict__ attn)             // (N*L, C) pre-projection output
{
    const int lane = threadIdx.x & 31;
    const int wave = threadIdx.x >> 5;
    const int unit = blockIdx.x * 8 + wave;      // (n*L + l)*M + m
    if (unit >= N_BATCH * N_QUERY * N_HEADS) return;
    const int m  = unit & (N_HEADS - 1);
    const int nl = unit >> 3;                    // n*L + l
    const int n  = nl / N_QUERY;

    // softmax over 16 (level,point) logits — redundantly per lane (broadcast loads)
    const float* wp = wgt_raw + (size_t)unit * 16;
    float wv[16];
    float mx = -3.4e38f;
#pragma unroll
    for (int j = 0; j < 16; ++j) { wv[j] = wp[j]; mx = fmaxf(mx, wv[j]); }
    float ssum = 0.0f;
#pragma unroll
    for (int j = 0; j < 16; ++j) { wv[j] = __expf(wv[j] - mx); ssum += wv[j]; }
    const float inv = 1.0f / ssum;

    const float* lp = locs + (size_t)nl * 256 + m * (N_POINTS * N_LEVELS * 2);
    const float* rp = refp + (size_t)nl * (N_LEVELS * 4);

    const int HW[4]     = {128, 64, 32, 16};
    const int starts[4] = {0, 16384, 20480, 21504};
    const float by  = (m + 0.5f) * (1.0f / N_HEADS);
    const int   cch = m * HEAD_DIM + lane;
    const float* Vn = value + (size_t)n * S_TOTAL * D_MODEL;

    float acc = 0.0f;
#pragma unroll
    for (int f = 0; f < N_LEVELS; ++f) {
        const float ww  = rp[f * 4 + 2], hh = rp[f * 4 + 3];
        const float tlx = rp[f * 4 + 0] - 0.5f * ww;
        const float tly = rp[f * 4 + 1] - 0.5f * hh;
        const int W = HW[f], H = HW[f];
        const float* V = Vn + (size_t)starts[f] * D_MODEL;
#pragma unroll
        for (int p = 0; p < N_POINTS; ++p) {
            const float lx = lp[(p * N_LEVELS + f) * 2 + 0];
            const float ly = lp[(p * N_LEVELS + f) * 2 + 1];
            const float bx = (p + 0.5f) * 0.25f;
            const float gx = (lx * ww * 0.25f + tlx + bx * ww) * 2.0f - 1.0f;
            const float gy = (ly * hh * 0.25f + tly + by * hh) * 2.0f - 1.0f;
            const float x  = (gx + 1.0f) * (W * 0.5f) - 0.5f;
            const float y  = (gy + 1.0f) * (H * 0.5f) - 0.5f;
            const float x0f = floorf(x), y0f = floorf(y);
            const int   ix0 = (int)x0f, iy0 = (int)y0f;
            const float fx  = x - x0f,  fy  = y - y0f;

            const bool xv0 = (ix0 >= 0)     && (ix0 < W);
            const bool xv1 = (ix0 + 1 >= 0) && (ix0 + 1 < W);
            const bool yv0 = (iy0 >= 0)     && (iy0 < H);
            const bool yv1 = (iy0 + 1 >= 0) && (iy0 + 1 < H);

            float v00 = 0.0f, v10 = 0.0f, v01 = 0.0f, v11 = 0.0f;
            if (yv0) {   // wave-uniform branches (x,y do not depend on lane)
                const float* Vr = V + (size_t)(iy0 * W) * D_MODEL;
                if (xv0) v00 = Vr[(size_t)ix0 * D_MODEL + cch];
                if (xv1) v10 = Vr[(size_t)(ix0 + 1) * D_MODEL + cch];
            }
            if (yv1) {
                const float* Vr = V + (size_t)((iy0 + 1) * W) * D_MODEL;
                if (xv0) v01 = Vr[(size_t)ix0 * D_MODEL + cch];
                if (xv1) v11 = Vr[(size_t)(ix0 + 1) * D_MODEL + cch];
            }
            const float bil = v00 * (1.0f - fx) * (1.0f - fy)
                            + v10 * fx * (1.0f - fy)
                            + v01 * (1.0f - fx) * fy
                            + v11 * fx * fy;
            acc += wv[f * 4 + p] * inv * bil;
        }
    }
    attn[(size_t)nl * D_MODEL + cch] = acc;
}

// ---------------------------------------------------------------------------
extern "C" void kernel_launch(void* const* d_in, const int* in_sizes, int n_in,
                              void* d_out, int out_size, void* d_ws, size_t ws_size,
                              hipStream_t stream)
{
    const float* query   = (const float*)d_in[0];
    const float* refp    = (const float*)d_in[1];
    const float* inflat  = (const float*)d_in[2];
    // d_in[3] spatial shapes, d_in[4] level starts: compile-time constants here
    const unsigned char* mask = (const unsigned char*)d_in[5];   // all-false
    const float* value_w = (const float*)d_in[6];
    const float* value_b = (const float*)d_in[7];
    const float* out_w   = (const float*)d_in[8];
    const float* out_b   = (const float*)d_in[9];
    const float* loc_w   = (const float*)d_in[10];
    const float* loc_b   = (const float*)d_in[11];
    const float* wgt_w   = (const float*)d_in[12];
    const float* wgt_b   = (const float*)d_in[13];

    const int NQ = N_BATCH * N_QUERY;   // 2400

    // workspace layout
    float* value   = (float*)d_ws;                                  // N*S*C
    float* wgt_raw = value   + (size_t)N_BATCH * S_TOTAL * D_MODEL; // NQ*128
    float* locs    = wgt_raw + (size_t)NQ * 128;                    // NQ*256
    float* attn    = locs    + (size_t)NQ * 256;                    // NQ*256

    // 1) value projection: (174080 x 256) x (256 x 256)^T + bias, masked
    {
        const int Mrows = N_BATCH * S_TOTAL;     // 174080 (multiple of 16)
        wmma_gemm_nt_kernel<<<Mrows / 16, 256, 0, stream>>>(
            inflat, value_w, value_b, value, Mrows, mask);
    }

    // 2) sampling-location + attention-weight projections
    {
        const int total = NQ * 384;
        qproj_kernel<<<(total + 255) / 256, 256, 0, stream>>>(
            query, wgt_w, wgt_b, loc_w, loc_b, wgt_raw, locs);
    }

    // 3) softmax + bilinear sampling + weighted sum -> attn (NQ x 256)
    msda_sample_kernel<<<(NQ * N_HEADS) / 8, 256, 0, stream>>>(
        value, wgt_raw, locs, refp, attn);

    // 4) output projection: (2400 x 256) x (256 x 256)^T + bias -> d_out
    {
        wmma_gemm_nt_kernel<<<NQ / 16, 256, 0, stream>>>(
            attn, out_w, out_b, (float*)d_out, NQ, nullptr);
    }
}